// TransformerMHSLayer_39779987096202
// MI455X (gfx1250) — compile-verified
//
#include <hip/hip_runtime.h>

#define S 256
#define H 12
#define E 64
#define KTOT 768
#define NL 25

typedef float v2f __attribute__((ext_vector_type(2)));
typedef float v8f __attribute__((ext_vector_type(8)));

// LDS layout (floats)
#define XT_PITCH 273                 // odd, 273 % 64 == 17 -> conflict-free stores & B reads
#define XT_SIZE  (64 * XT_PITCH)     // 17472
#define WC_PITCH 68                  // %4==0 (float4 stores), %64==4 -> conflict-free A reads
#define WC_SIZE  (32 * WC_PITCH)     // 2176
#define P_OFF    (XT_SIZE + WC_SIZE) // 256 floats of p
#define V_OFF    (P_OFF + S)         // 64 floats of v
#define B_OFF    (V_OFF + E)         // 32 floats of bias
#define SMEM_FLOATS (B_OFF + 32)     // 20000 floats = 80000 bytes

__global__ __launch_bounds__(256) void
fused_rel_attn_proj_kernel(const float* __restrict__ ap,   // (B,H,S,S)
                           const float* __restrict__ va,   // (B,H,S,E)
                           const float* __restrict__ rel,  // (B,S,S,768)
                           const float* __restrict__ W,    // (25,768)
                           const float* __restrict__ bias, // (25,)
                           float* __restrict__ out)        // (B,S,25,S)
{
    extern __shared__ float smem[];
    float* xT = smem;            // [64][XT_PITCH]  x^T chunk for current head
    float* Wc = smem + XT_SIZE;  // [32][WC_PITCH]  zero-padded W chunk
    float* pb = smem + P_OFF;    // p[b,h,i,j] for i=0..255
    float* vb = smem + V_OFF;    // v[b,h,j,e] for e=0..63
    float* bb = smem + B_OFF;    // bias (padded to 32)

    const int t    = threadIdx.x;
    const int b    = blockIdx.x >> 8;
    const int j    = blockIdx.x & (S - 1);
    const int wave = t >> 5;
    const int lane = t & 31;
    const int lm   = lane & 15;   // N / M-within-tile lane index
    const int kh   = lane >> 4;   // half-wave selector (K offset pairs)
    const int i0   = wave * 32;   // this wave's first i (2 i-tiles of 16)

    if (t < 32) bb[t] = (t < NL) ? bias[t] : 0.0f;

    v8f acc[2][2];                // [l-tile][i-tile]
#pragma unroll
    for (int a = 0; a < 2; ++a)
#pragma unroll
        for (int c = 0; c < 2; ++c)
#pragma unroll
            for (int q = 0; q < 8; ++q) acc[a][c][q] = 0.0f;

    for (int h = 0; h < H; ++h) {
        __syncthreads();  // previous head's LDS readers done

        // ---- Phase A: stage p, v, W-chunk into LDS ----
        pb[t] = ap[(((size_t)b * H + h) * S + t) * S + j];
        if (t < E) vb[t] = va[(((size_t)b * H + h) * S + j) * E + t];
        {
            const int l = t >> 3;          // 0..31
            const int e = (t & 7) * 8;     // 0,8,...,56
            float4 w0 = {0, 0, 0, 0}, w1 = {0, 0, 0, 0};
            if (l < NL) {
                const float* wp = W + (size_t)l * KTOT + h * E + e;
                w0 = *(const float4*)wp;
                w1 = *(const float4*)(wp + 4);
            }
            float* d = Wc + l * WC_PITCH + e;
            *(float4*)d       = w0;
            *(float4*)(d + 4) = w1;
        }
        __syncthreads();

        // ---- Phase B: stream rel chunk, fuse p*(v+rel), transpose into xT ----
        {
            const int e4   = (t & 15) * 4;   // 0,4,...,60
            const int isub = t >> 4;         // 0..15
            const float vv0 = vb[e4 + 0], vv1 = vb[e4 + 1];
            const float vv2 = vb[e4 + 2], vv3 = vb[e4 + 3];
#pragma unroll
            for (int pass = 0; pass < 16; ++pass) {
                const int i = pass * 16 + isub;
                const float4 r4 = *(const float4*)&rel[(((size_t)b * S + i) * S + j) * (size_t)KTOT + h * E + e4];
                const float pv = pb[i];
                xT[(e4 + 0) * XT_PITCH + i] = pv * (vv0 + r4.x);
                xT[(e4 + 1) * XT_PITCH + i] = pv * (vv1 + r4.y);
                xT[(e4 + 2) * XT_PITCH + i] = pv * (vv2 + r4.z);
                xT[(e4 + 3) * XT_PITCH + i] = pv * (vv3 + r4.w);
            }
        }
        __syncthreads();

        // ---- Phase C: WMMA GEMM over this head's K=64 (16 steps of K=4) ----
        // A (16x4 f32, M=labels): lane: M=lm, VGPR0 K=k0+2*kh, VGPR1 K=k0+2*kh+1
        // B (4x16 f32, N=i):      lane: N=lm, VGPR0 K=k0+2*kh, VGPR1 K=k0+2*kh+1
#pragma unroll
        for (int kk = 0; kk < E; kk += 4) {
            const int kA = kk + 2 * kh;
            v2f a0 = *(const v2f*)&Wc[lm * WC_PITCH + kA];
            v2f a1 = *(const v2f*)&Wc[(16 + lm) * WC_PITCH + kA];
            v2f b0, b1;
            b0.x = xT[kA * XT_PITCH + i0 + lm];
            b0.y = xT[(kA + 1) * XT_PITCH + i0 + lm];
            b1.x = xT[kA * XT_PITCH + i0 + 16 + lm];
            b1.y = xT[(kA + 1) * XT_PITCH + i0 + 16 + lm];
            acc[0][0] = __builtin_amdgcn_wmma_f32_16x16x4_f32(false, a0, false, b0, (short)0, acc[0][0], false, false);
            acc[1][0] = __builtin_amdgcn_wmma_f32_16x16x4_f32(false, a1, false, b0, (short)0, acc[1][0], false, false);
            acc[0][1] = __builtin_amdgcn_wmma_f32_16x16x4_f32(false, a0, false, b1, (short)0, acc[0][1], false, false);
            acc[1][1] = __builtin_amdgcn_wmma_f32_16x16x4_f32(false, a1, false, b1, (short)0, acc[1][1], false, false);
        }
    }

    // ---- Epilogue: add bias, store O[b,j,l,i] (contiguous in i per half-wave) ----
#pragma unroll
    for (int lt = 0; lt < 2; ++lt) {
#pragma unroll
        for (int it = 0; it < 2; ++it) {
            const int i = i0 + it * 16 + lm;
#pragma unroll
            for (int r = 0; r < 8; ++r) {
                const int l = lt * 16 + r + 8 * kh;  // C/D layout: VGPR r -> M=r (lanes 0-15), M=r+8 (16-31)
                if (l < NL) {
                    out[(((size_t)b * S + j) * NL + l) * S + i] = acc[lt][it][r] + bb[l];
                }
            }
        }
    }
}

extern "C" void kernel_launch(void* const* d_in, const int* in_sizes, int n_in,
                              void* d_out, int out_size, void* d_ws, size_t ws_size,
                              hipStream_t stream) {
    const float* ap   = (const float*)d_in[0];  // attention_probs (2,12,256,256)
    const float* va   = (const float*)d_in[1];  // value_attentions (2,12,256,64)
    const float* rel  = (const float*)d_in[2];  // relative_positions (2,256,256,768)
    const float* W    = (const float*)d_in[3];  // (25,768)
    const float* bias = (const float*)d_in[4];  // (25,)
    float* out = (float*)d_out;                 // (2,256,25,256)

    dim3 grid(2 * S);   // one block per (b, j)
    dim3 block(256);    // 8 wave32
    size_t smem = SMEM_FLOATS * sizeof(float);
    hipLaunchKernelGGL(fused_rel_attn_proj_kernel, grid, block, smem, stream,
                       ap, va, rel, W, bias, out);
}